// MultiHeadAttention_27487790694999
// MI455X (gfx1250) — compile-verified
//
#include <hip/hip_runtime.h>

typedef __attribute__((ext_vector_type(2))) float v2f;
typedef __attribute__((ext_vector_type(8))) float v8f;

constexpr int EMB   = 128;
constexpr int SEQ   = 2048;
constexpr int BATCH = 4;
constexpr int HEADS = 16;
constexpr int HD    = 8;                       // head dim
constexpr float ATT_SCALE = 0.35355339059327373f; // 8^-0.5

// D = A(16x4) * B(4x16) + C, fp32 WMMA.
// A: lane&15 = row M, lane>=16 selects K-pair {2,3} vs {0,1}, float2/lane.
// B: lane&15 = col N, lane>=16 selects K-pair {2,3} vs {0,1}, float2/lane (column-major per lane).
// C/D: 8 VGPRs, element (M = r + 8*(lane>>4), N = lane&15).
__device__ __forceinline__ v8f wmma4_f32(v2f a, v2f b, v8f c) {
  return __builtin_amdgcn_wmma_f32_16x16x4_f32(
      /*neg_a=*/false, a, /*neg_b=*/false, b,
      /*c_mod=*/(short)0, c, /*reuse_a=*/false, /*reuse_b=*/false);
}

// out[M,N] = sum_k in[M,k] * W[N,k] + bias[N]   (torch Linear: x @ W.T + b)
// M = BATCH*SEQ = 8192 rows, N = K = EMB = 128. One wave per 16x16 tile.
__global__ __launch_bounds__(128) void linear_wmma(
    const float* __restrict__ x, const float* __restrict__ W,
    const float* __restrict__ bias, float* __restrict__ out) {
  const int lane = threadIdx.x & 31;
  const int wave = threadIdx.x >> 5;
  const int tile = blockIdx.x * 4 + wave;   // 4096 tiles total
  const int mt = tile >> 3;                 // 512 M-tiles
  const int nt = tile & 7;                  // 8 N-tiles
  const int m0 = mt * 16, n0 = nt * 16;
  const int lo = lane & 15, hi = lane >> 4;

  const float* arow = x + (size_t)(m0 + lo) * EMB + 2 * hi;  // A: row of x
  const float* brow = W + (size_t)(n0 + lo) * EMB + 2 * hi;  // B col n = W row n (W.T)
  v8f c = {};
#pragma unroll
  for (int k0 = 0; k0 < EMB; k0 += 4) {
    v2f a = *(const v2f*)(arow + k0);
    v2f b = *(const v2f*)(brow + k0);
    c = wmma4_f32(a, b, c);
  }
  const float bn = bias[n0 + lo];
  float* orow = out + (size_t)(m0 + 8 * hi) * EMB + n0 + lo;
#pragma unroll
  for (int r = 0; r < 8; ++r)
    orow[(size_t)r * EMB] = c[r] + bn;
}

// Fused causal sigmoid-attention, streaming over key tiles.
// One wave per (b, h, q-tile). acc holds [Sum w*V | Sum w] (col 8 = rowsum).
__global__ __launch_bounds__(128) void attn_wmma(
    const float* __restrict__ Q, const float* __restrict__ K,
    const float* __restrict__ V, float* __restrict__ O) {
  __shared__ float wlds[4 * 16 * 17];       // per-wave 16x17 staging (transpose W-tile)
  const int lane = threadIdx.x & 31;
  const int wave = threadIdx.x >> 5;
  const int g  = blockIdx.x * 4 + wave;     // 8192 wave-tiles
  const int qt = g & 127;                   // SEQ/16
  const int bh = g >> 7;
  const int b = bh >> 4, h = bh & 15;
  const int lo = lane & 15, hi = lane >> 4;
  const size_t base = (size_t)b * SEQ * EMB + (size_t)h * HD;
  const float* Qb = Q + base;
  const float* Kb = K + base;
  const float* Vb = V + base;
  float*       Ob = O + base;
  const int q0 = qt * 16;

  // Persistent Q A-fragments (K-dim = head dim = 8 -> two 16x16x4 steps)
  const float* qrow = Qb + (size_t)(q0 + lo) * EMB + 2 * hi;
  const v2f aq0 = *(const v2f*)(qrow);
  const v2f aq1 = *(const v2f*)(qrow + 4);

  float* wl = wlds + wave * (16 * 17);
  v8f acc = {};

  for (int kt = 0; kt <= qt; ++kt) {        // causal: only key tiles <= query tile
    const int k0r = kt * 16;

    // ---- scores tile: S = Q_tile @ K_tile^T  (two K=4 WMMAs) ----
    const float* krow = Kb + (size_t)(k0r + lo) * EMB + 2 * hi; // B col n = key row k0r+n
    const v2f bk0 = *(const v2f*)(krow);
    const v2f bk1 = *(const v2f*)(krow + 4);
    v8f s = {};
    s = wmma4_f32(aq0, bk0, s);
    s = wmma4_f32(aq1, bk1, s);

    // ---- mask + sigmoid, stage W-tile to LDS (D-layout -> A-layout transpose) ----
    const int n = k0r + lo;
#pragma unroll
    for (int r = 0; r < 8; ++r) {
      const int m = q0 + r + 8 * hi;
      const float w = (n <= m) ? 1.0f / (1.0f + __expf(-ATT_SCALE * s[r])) : 0.0f;
      wl[(r + 8 * hi) * 17 + lo] = w;
    }
    asm volatile("s_wait_dscnt 0" ::: "memory");  // RAW: LDS stores -> LDS reads

    // ---- acc += W_tile(16x16) @ [V_tile | 1 | 0...](16x16), 4 K=4 WMMAs ----
#pragma unroll
    for (int k0 = 0; k0 < 16; k0 += 4) {
      v2f a;
      a.x = wl[lo * 17 + k0 + 2 * hi];
      a.y = wl[lo * 17 + k0 + 2 * hi + 1];
      const int kk = k0r + k0 + 2 * hi;           // global key row for this K-pair
      const float* vp = Vb + (size_t)kk * EMB + (lo & 7);
      const float v0 = vp[0];
      const float v1 = vp[EMB];
      v2f bb;
      bb.x = (lo < 8) ? v0 : ((lo == 8) ? 1.0f : 0.0f);  // col 8 = ones -> rowsum
      bb.y = (lo < 8) ? v1 : ((lo == 8) ? 1.0f : 0.0f);
      acc = wmma4_f32(a, bb, acc);
    }
  }

  // ---- normalize by rowsum (accumulator column 8: lanes 8 / 24) and store ----
#pragma unroll
  for (int r = 0; r < 8; ++r) {
    const float rs = __shfl(acc[r], (lane & 16) | 8, 32);
    acc[r] = acc[r] / rs;
  }
  if (lo < 8) {
    float* orow = Ob + (size_t)(q0 + 8 * hi) * EMB + lo;
#pragma unroll
    for (int r = 0; r < 8; ++r)
      orow[(size_t)r * EMB] = acc[r];
  }
}

extern "C" void kernel_launch(void* const* d_in, const int* in_sizes, int n_in,
                              void* d_out, int out_size, void* d_ws, size_t ws_size,
                              hipStream_t stream) {
  const float* x  = (const float*)d_in[0];
  const float* Wq = (const float*)d_in[1];
  const float* bq = (const float*)d_in[2];
  const float* Wk = (const float*)d_in[3];
  const float* bk = (const float*)d_in[4];
  const float* Wv = (const float*)d_in[5];
  const float* bv = (const float*)d_in[6];
  const float* Wo = (const float*)d_in[7];
  const float* bo = (const float*)d_in[8];
  float* out = (float*)d_out;

  const size_t mat = (size_t)BATCH * SEQ * EMB;  // 1M floats = 4 MB each
  float* Qw = (float*)d_ws;                      // needs 16 MB of d_ws total
  float* Kw = Qw + mat;
  float* Vw = Kw + mat;
  float* Aw = Vw + mat;

  dim3 blk(128);           // 4 waves (wave32) per block
  dim3 gproj(1024);        // 4096 16x16 tiles / 4 waves
  linear_wmma<<<gproj, blk, 0, stream>>>(x, Wq, bq, Qw);
  linear_wmma<<<gproj, blk, 0, stream>>>(x, Wk, bk, Kw);
  linear_wmma<<<gproj, blk, 0, stream>>>(x, Wv, bv, Vw);
  attn_wmma<<<dim3(2048), blk, 0, stream>>>(Qw, Kw, Vw, Aw);  // 8192 (b,h,qt) wave-tiles
  linear_wmma<<<gproj, blk, 0, stream>>>(Aw, Wo, bo, out);
}